// GCN_W_Global_14491219657228
// MI455X (gfx1250) — compile-verified
//
#include <hip/hip_runtime.h>
#include <math.h>

typedef float v2f __attribute__((ext_vector_type(2)));
typedef float v8f __attribute__((ext_vector_type(8)));

__device__ __forceinline__ float gelu_exact(float x) {
    // exact gelu: 0.5*x*(1+erf(x/sqrt(2)))
    return 0.5f * x * (1.0f + erff(x * 0.70710678118654752440f));
}

// ---------------------------------------------------------------- tiny prep
__global__ void k_softmax3(const float* __restrict__ mw, float* __restrict__ sp) {
    if (blockIdx.x == 0 && threadIdx.x == 0) {
        float m  = fmaxf(mw[0], fmaxf(mw[1], mw[2]));
        float e0 = expf(mw[0] - m), e1 = expf(mw[1] - m), e2 = expf(mw[2] - m);
        float inv = 1.0f / (e0 + e1 + e2);
        sp[0] = e0 * inv; sp[1] = e1 * inv; sp[2] = e2 * inv;
    }
}

__global__ void k_fill(float* __restrict__ p, float v, int n) {
    int i = blockIdx.x * blockDim.x + threadIdx.x;
    if (i < n) p[i] = v;
}

__global__ void k_ew_set(const int* __restrict__ mask, int n,
                         const float* __restrict__ sp, int which,
                         float* __restrict__ ew) {
    int i = blockIdx.x * blockDim.x + threadIdx.x;
    if (i < n) ew[mask[i]] = sp[which];
}

// pack W_gcn (64x61) into WMMA-friendly pairs: Wp[kb][kh2][n][2], zero-padded n>=61
__global__ void k_pack_w(const float* __restrict__ Wg, float* __restrict__ Wp) {
    int t = blockIdx.x * blockDim.x + threadIdx.x;   // 16*2*64 = 2048
    if (t >= 2048) return;
    int nc  = t & 63;
    int kh2 = (t >> 6) & 1;
    int kb  = t >> 7;
    int k0  = kb * 4 + kh2 * 2;
    Wp[t * 2]     = (nc < 61) ? Wg[k0 * 61 + nc]       : 0.0f;
    Wp[t * 2 + 1] = (nc < 61) ? Wg[(k0 + 1) * 61 + nc] : 0.0f;
}

// ------------------------------------------------- node embedding -> h (N,64)
__global__ void k_embed(const float* __restrict__ x, const float* __restrict__ Wemb,
                        const float* __restrict__ bemb, float* __restrict__ h, int N) {
    int tid = blockIdx.x * blockDim.x + threadIdx.x;
    int n = tid >> 4, c4 = (tid & 15) * 4;
    if (n >= N) return;
    float x0 = x[n * 12 + 0], x1 = x[n * 12 + 1];
    float r[4];
#pragma unroll
    for (int i = 0; i < 4; ++i) {
        int c = c4 + i;
        r[i] = (c < 54) ? fmaf(x0, Wemb[c], fmaf(x1, Wemb[54 + c], bemb[c]))
                        : x[n * 12 + (c - 52)];   // x[:,2:] passthrough
    }
    *(float4*)(h + (size_t)n * 64 + c4) = make_float4(r[0], r[1], r[2], r[3]);
}

// ------------------------------------- WMMA GEMM: hg(N,64pad) = h(N,64) @ Wp
__global__ void k_gemm_gcn(const float* __restrict__ h, const float* __restrict__ Wp,
                           float* __restrict__ hg, int N, int n_mtiles) {
    int wave = threadIdx.x >> 5;
    int lane = threadIdx.x & 31;
    int mt = blockIdx.x * (blockDim.x >> 5) + wave;
    if (mt >= n_mtiles) return;               // uniform per wave; EXEC stays full
    const int m0  = mt * 16;
    const int kh2 = lane >> 4;                // lanes 16-31 hold K+2,K+3
    int mrow = m0 + (lane & 15);
    if (mrow >= N) mrow = N - 1;              // clamp loads for ragged last tile
    const int ncl = lane & 15;

    for (int nt = 0; nt < 4; ++nt) {
        v8f acc = {};
        const int nc = nt * 16 + ncl;
        for (int kb = 0; kb < 16; ++kb) {
            const int k0 = kb * 4 + kh2 * 2;
            v2f a = *(const v2f*)(h + (size_t)mrow * 64 + k0);
            v2f b = *(const v2f*)(Wp + (((kb * 2 + kh2) * 64) + nc) * 2);
            acc = __builtin_amdgcn_wmma_f32_16x16x4_f32(
                false, a, false, b, (short)0, acc, false, false);
        }
        const int mb = m0 + kh2 * 8;
        if (m0 + 16 <= N) {                    // uniform fast path: no per-row guards
#pragma unroll
            for (int i = 0; i < 8; ++i)
                hg[(size_t)(mb + i) * 64 + nc] = acc[i];
        } else {
#pragma unroll
            for (int i = 0; i < 8; ++i)
                if (mb + i < N) hg[(size_t)(mb + i) * 64 + nc] = acc[i];
        }
    }
}

// --------------------------------------------------------- degree + normalize
__global__ void k_deg_edges(const int* __restrict__ ei, const float* __restrict__ ew,
                            float* __restrict__ deg, int E) {
    int e = blockIdx.x * blockDim.x + threadIdx.x;
    if (e < E) atomicAdd(&deg[ei[E + e]], ew[e]);   // col = ei[1][e]
}

__global__ void k_dis(float* __restrict__ deg, int N) {
    int n = blockIdx.x * blockDim.x + threadIdx.x;
    if (n >= N) return;
    float d = deg[n];
    deg[n] = (d > 0.0f) ? rsqrtf(fmaxf(d, 1e-12f)) : 0.0f;   // in place -> dis
}

// self-loop term: acc[n,:] = dis[n]^2 * hg[n,:]  (cols 61..63 are zero anyway)
__global__ void k_acc_self(const float* __restrict__ hg, const float* __restrict__ dis,
                           float* __restrict__ acc, int N) {
    int tid = blockIdx.x * blockDim.x + threadIdx.x;
    int n = tid >> 4, c4 = (tid & 15) * 4;
    if (n >= N) return;
    float d = dis[n];
    float s = d * d;
    float4 v = *(const float4*)(hg + (size_t)n * 64 + c4);
    v.x *= s; v.y *= s; v.z *= s; v.w *= s;
    *(float4*)(acc + (size_t)n * 64 + c4) = v;
}

// edge scatter: acc[col] += dis[row]*ew*dis[col] * hg[row]   (16 threads/edge, f4)
__global__ void k_edge_scatter(const int* __restrict__ ei, const float* __restrict__ ew,
                               const float* __restrict__ dis, const float* __restrict__ hg,
                               float* __restrict__ acc, int E) {
    long long tid = (long long)blockIdx.x * blockDim.x + threadIdx.x;
    int e = (int)(tid >> 4), c4 = (int)(tid & 15) * 4;
    if (e >= E) return;
    int r  = ei[e];
    int cl = ei[E + e];
    float norm = dis[r] * ew[e] * dis[cl];
    float4 v = *(const float4*)(hg + (size_t)r * 64 + c4);
    float* dst = acc + (size_t)cl * 64 + c4;
    if (c4 < 60) {
        atomicAdd(dst + 0, norm * v.x);
        atomicAdd(dst + 1, norm * v.y);
        atomicAdd(dst + 2, norm * v.z);
        atomicAdd(dst + 3, norm * v.w);
    } else {                                   // last quad: only col 60 is live
        atomicAdd(dst + 0, norm * v.x);
    }
}

// -------------------------------------------- fn-MLP writes z[:,61:64] directly
__global__ void k_fn(const float* __restrict__ fa, const float* __restrict__ ir,
                     const float* __restrict__ W1, const float* __restrict__ b1,
                     const float* __restrict__ W2, const float* __restrict__ b2,
                     float* __restrict__ z, int G) {
    int g = blockIdx.x * blockDim.x + threadIdx.x;
    if (g >= G) return;
    float e0 = fa[g * 2], e1 = fa[g * 2 + 1], e2 = ir[g];
    float t[16];
#pragma unroll
    for (int j = 0; j < 16; ++j)
        t[j] = gelu_exact(e0 * W1[j] + e1 * W1[16 + j] + e2 * W1[32 + j] + b1[j]);
#pragma unroll
    for (int o = 0; o < 3; ++o) {
        float s = b2[o];
#pragma unroll
        for (int j = 0; j < 16; ++j) s += t[j] * W2[j * 3 + o];
        z[g * 64 + 61 + o] = s;
    }
}

// gelu(acc + b_gcn) then global_add_pool into z[:,0:61]   (16 threads/node, f4)
__global__ void k_pool(const float* __restrict__ acc, const float* __restrict__ bgcn,
                       const int* __restrict__ batch, float* __restrict__ z, int N) {
    int tid = blockIdx.x * blockDim.x + threadIdx.x;
    int n = tid >> 4, c4 = (tid & 15) * 4;
    if (n >= N) return;
    int g = batch[n];
    float4 v = *(const float4*)(acc + (size_t)n * 64 + c4);
    float* dst = z + (size_t)g * 64 + c4;
    if (c4 < 60) {
        atomicAdd(dst + 0, gelu_exact(v.x + bgcn[c4 + 0]));
        atomicAdd(dst + 1, gelu_exact(v.y + bgcn[c4 + 1]));
        atomicAdd(dst + 2, gelu_exact(v.z + bgcn[c4 + 2]));
        atomicAdd(dst + 3, gelu_exact(v.w + bgcn[c4 + 3]));
    } else {
        atomicAdd(dst + 0, gelu_exact(v.x + bgcn[60]));
    }
}

// -------------------- WMMA backbone: z2 = gelu(z(G,64) @ Wbb(64,64) + b_bb1)
__global__ void k_gemm_bb(const float* __restrict__ z, const float* __restrict__ W,
                          const float* __restrict__ bias, float* __restrict__ z2,
                          int n_mtiles) {
    int wave = threadIdx.x >> 5;
    int lane = threadIdx.x & 31;
    int mt = blockIdx.x * (blockDim.x >> 5) + wave;
    if (mt >= n_mtiles) return;
    const int m0    = mt * 16;
    const int khalf = (lane >> 4) * 2;
    const int mrow  = m0 + (lane & 15);
    const int ncl   = lane & 15;

    for (int nt = 0; nt < 4; ++nt) {
        v8f acc = {};
        const int nc = nt * 16 + ncl;
        for (int kb = 0; kb < 16; ++kb) {
            const int k0 = kb * 4 + khalf;
            v2f a = *(const v2f*)(z + (size_t)mrow * 64 + k0);
            v2f b; b.x = W[k0 * 64 + nc]; b.y = W[(k0 + 1) * 64 + nc];
            acc = __builtin_amdgcn_wmma_f32_16x16x4_f32(
                false, a, false, b, (short)0, acc, false, false);
        }
        const int mb = m0 + ((lane >> 4) * 8);
        const float bv = bias[nc];
#pragma unroll
        for (int i = 0; i < 8; ++i)
            z2[(size_t)(mb + i) * 64 + nc] = gelu_exact(acc[i] + bv);
    }
}

__global__ void k_logits(const float* __restrict__ z2, const float* __restrict__ Wout,
                         const float* __restrict__ bout, float* __restrict__ out, int G) {
    int g = blockIdx.x * blockDim.x + threadIdx.x;
    if (g >= G) return;
    float s = bout[0];
#pragma unroll
    for (int k = 0; k < 64; ++k) s += z2[g * 64 + k] * Wout[k];
    out[g] = s;
}

static inline int cdiv(long long a, long long b) { return (int)((a + b - 1) / b); }

extern "C" void kernel_launch(void* const* d_in, const int* in_sizes, int n_in,
                              void* d_out, int out_size, void* d_ws, size_t ws_size,
                              hipStream_t stream) {
    const float* x      = (const float*)d_in[0];
    const int*   ei     = (const int*)  d_in[1];
    const int*   batch  = (const int*)  d_in[2];
    const int*   km     = (const int*)  d_in[3];
    const int*   um     = (const int*)  d_in[4];
    const int*   om     = (const int*)  d_in[5];
    const float* fa     = (const float*)d_in[6];
    const float* ir     = (const float*)d_in[7];
    const float* mw     = (const float*)d_in[8];
    const float* Wemb   = (const float*)d_in[9];
    const float* bemb   = (const float*)d_in[10];
    const float* Wgcn   = (const float*)d_in[11];
    const float* bgcn   = (const float*)d_in[12];
    const float* Wfn1   = (const float*)d_in[13];
    const float* bfn1   = (const float*)d_in[14];
    const float* Wfn2   = (const float*)d_in[15];
    const float* bfn2   = (const float*)d_in[16];
    const float* Wbb1   = (const float*)d_in[17];
    const float* bbb1   = (const float*)d_in[18];
    const float* Wout   = (const float*)d_in[19];
    const float* bout   = (const float*)d_in[20];
    float* out = (float*)d_out;

    const int N  = in_sizes[2];
    const int E  = in_sizes[1] / 2;
    const int G  = in_sizes[7];
    const int nk = in_sizes[3], nu = in_sizes[4], no = in_sizes[5];

    // workspace layout (floats)
    float* ws  = (float*)d_ws;
    float* sp  = ws;                      // 16
    float* Wp  = sp + 16;                 // 4096 packed W_gcn
    float* ewp = Wp + 4096;               // E
    float* h   = ewp + E;                 // N*64  (reused as acc after GEMM)
    float* hg  = h + (size_t)N * 64;      // N*64
    float* deg = hg + (size_t)N * 64;     // N (becomes dis in place)
    float* z   = deg + N;                 // G*64
    float* z2  = z + (size_t)G * 64;      // G*64

    const int B = 256;

    // 1) softmax of msg weights + pack GCN weights for WMMA
    k_softmax3<<<1, 32, 0, stream>>>(mw, sp);
    k_pack_w<<<cdiv(2048, B), B, 0, stream>>>(Wgcn, Wp);
    // 2) edge weights = 1, then mask writes (launch order = torch precedence)
    k_fill<<<cdiv(E, B), B, 0, stream>>>(ewp, 1.0f, E);
    k_ew_set<<<cdiv(nk, B), B, 0, stream>>>(km, nk, sp, 0, ewp);
    k_ew_set<<<cdiv(nu, B), B, 0, stream>>>(um, nu, sp, 1, ewp);
    k_ew_set<<<cdiv(no, B), B, 0, stream>>>(om, no, sp, 2, ewp);
    // 3) node embedding -> h (N,64)
    k_embed<<<cdiv((long long)N * 16, B), B, 0, stream>>>(x, Wemb, bemb, h, N);
    // 4) hg = h @ W_gcn   (f32 WMMA, packed/zero-padded B operand)
    const int mtiles = cdiv(N, 16);
    k_gemm_gcn<<<cdiv(mtiles, 8), B, 0, stream>>>(h, Wp, hg, N, mtiles);
    // 5) weighted degree (self loop contributes 1.0) -> dis = rsqrt(deg)
    k_fill<<<cdiv(N, B), B, 0, stream>>>(deg, 1.0f, N);
    k_deg_edges<<<cdiv(E, B), B, 0, stream>>>(ei, ewp, deg, E);
    k_dis<<<cdiv(N, B), B, 0, stream>>>(deg, N);
    // 6) message passing: self-loop init then edge scatter (h reused as acc)
    float* acc = h;
    k_acc_self<<<cdiv((long long)N * 16, B), B, 0, stream>>>(hg, deg, acc, N);
    k_edge_scatter<<<cdiv((long long)E * 16, B), B, 0, stream>>>(ei, ewp, deg, hg, acc, E);
    // 7) z = [pool(gelu(acc+b_gcn)), fn_mlp]
    k_fill<<<cdiv(G * 64, B), B, 0, stream>>>(z, 0.0f, G * 64);
    k_fn<<<cdiv(G, B), B, 0, stream>>>(fa, ir, Wfn1, bfn1, Wfn2, bfn2, z, G);
    k_pool<<<cdiv((long long)N * 16, B), B, 0, stream>>>(acc, bgcn, batch, z, N);
    // 8) backbone GEMM (f32 WMMA) + gelu, then output head
    k_gemm_bb<<<cdiv(G / 16, 8), B, 0, stream>>>(z, Wbb1, bbb1, z2, G / 16);
    k_logits<<<cdiv(G, B), B, 0, stream>>>(z2, Wout, bout, out, G);
}